// GCN_86354612453593
// MI455X (gfx1250) — compile-verified
//
#include <hip/hip_runtime.h>

typedef __attribute__((ext_vector_type(2))) float v2f;
typedef __attribute__((ext_vector_type(8))) float v8f;

#define N_NODES 100000
#define N_EDGES 3200000
#define IN_CH 512
#define HID 16

// ---------------- init: deg = 1.0 (self loop), agg = 0 ----------------
__global__ void gcn_init(float* __restrict__ deg, float* __restrict__ agg) {
    int i = blockIdx.x * blockDim.x + threadIdx.x;
    if (i < N_NODES * HID) agg[i] = 0.0f;
    if (i < N_NODES) deg[i] = 1.0f;
}

// ---------------- degree count over targets ----------------
__global__ void gcn_degree(const int* __restrict__ col, float* __restrict__ deg) {
    int e = blockIdx.x * blockDim.x + threadIdx.x;
    if (e < N_EDGES) atomicAdd(&deg[col[e]], 1.0f);
}

// ---------------- deg -> rsqrt(deg) in place ----------------
__global__ void gcn_rsqrt(float* __restrict__ deg) {
    int i = blockIdx.x * blockDim.x + threadIdx.x;
    if (i < N_NODES) deg[i] = rsqrtf(deg[i]);
}

// ---------------- xw = x @ W1 via V_WMMA_F32_16X16X4_F32 ----------------
// One wave computes one 16-row x 16-col tile; K consumed 4 at a time.
// A layout (32-bit A 16x4): lanes 0-15 = K0/K1 in the 2 VGPRs, lanes 16-31 = K2/K3.
// B layout (4x16): VGPR0 = rows K0 (lanes 0-15) / K2 (lanes 16-31), VGPR1 = K1/K3.
// D layout: VGPR r -> row r (lanes 0-15), row r+8 (lanes 16-31), col = lane&15.
__global__ void __launch_bounds__(256) gcn_xw_wmma(const float* __restrict__ x,
                                                   const float* __restrict__ W1,
                                                   float* __restrict__ xw) {
    __shared__ float sW1[IN_CH * HID];  // 32 KB of the 320 KB WGP LDS
    for (int i = threadIdx.x; i < (IN_CH * HID) / 4; i += 256)
        ((float4*)sW1)[i] = ((const float4*)W1)[i];
    __syncthreads();

    const int wave = threadIdx.x >> 5;
    const int lane = threadIdx.x & 31;
    const int half = lane >> 4;   // 0: K pair {0,1}, 1: K pair {2,3}
    const int lrow = lane & 15;
    const int tile = blockIdx.x * 8 + wave;
    if (tile * 16 >= N_NODES) return;  // wave-uniform; EXEC stays all-ones

    const float* xrow = x + (size_t)(tile * 16 + lrow) * IN_CH + 2 * half;
    const float* bptr = sW1 + (2 * half) * HID + lrow;

    v8f acc = {};
#pragma unroll 8
    for (int k0 = 0; k0 < IN_CH; k0 += 4) {
        v2f a = *(const v2f*)(xrow + k0);          // x[row][k0+2h .. k0+2h+1]
        v2f b;
        b.x = bptr[k0 * HID];                      // W1[k0+2h  ][lrow]
        b.y = bptr[k0 * HID + HID];                // W1[k0+2h+1][lrow]
        acc = __builtin_amdgcn_wmma_f32_16x16x4_f32(
            /*neg_a=*/false, a, /*neg_b=*/false, b,
            /*c_mod=*/(short)0, acc, /*reuse_a=*/false, /*reuse_b=*/false);
    }

    float* o = xw + (size_t)(tile * 16 + 8 * half) * HID + lrow;
#pragma unroll
    for (int r = 0; r < 8; ++r) o[r * HID] = acc[r];
}

// ---------------- edge scatter: 16 lanes per edge, coalesced atomics ----------------
__global__ void gcn_scatter(const int* __restrict__ row, const int* __restrict__ col,
                            const float* __restrict__ dinv, const float* __restrict__ xw,
                            float* __restrict__ agg) {
    long long t = (long long)blockIdx.x * blockDim.x + threadIdx.x;
    int e = (int)(t >> 4);
    int c = (int)t & 15;
    if (e < N_EDGES) {
        int r = row[e];
        int d = col[e];
        float nrm = dinv[r] * dinv[d];
        atomicAdd(&agg[(size_t)d * HID + c], xw[(size_t)r * HID + c] * nrm);
    }
}

// ---------------- finalize: add self-loop + b1, apply 16->2 linear ----------------
__global__ void gcn_out(const float* __restrict__ agg, const float* __restrict__ xw,
                        const float* __restrict__ dinv, const float* __restrict__ b1,
                        const float* __restrict__ W2, const float* __restrict__ b2,
                        float* __restrict__ out) {
    int i = blockIdx.x * blockDim.x + threadIdx.x;
    if (i >= N_NODES) return;
    float s = dinv[i] * dinv[i];  // self-loop norm
    float o0 = b2[0], o1 = b2[1];
#pragma unroll
    for (int c = 0; c < HID; ++c) {
        float h = agg[(size_t)i * HID + c] + xw[(size_t)i * HID + c] * s + b1[c];
        o0 += h * W2[c * 2 + 0];
        o1 += h * W2[c * 2 + 1];
    }
    out[i * 2 + 0] = o0;
    out[i * 2 + 1] = o1;
}

extern "C" void kernel_launch(void* const* d_in, const int* in_sizes, int n_in,
                              void* d_out, int out_size, void* d_ws, size_t ws_size,
                              hipStream_t stream) {
    const float* x  = (const float*)d_in[0];
    const int*   ei = (const int*)d_in[1];
    const float* W1 = (const float*)d_in[2];
    const float* b1 = (const float*)d_in[3];
    const float* W2 = (const float*)d_in[4];
    const float* b2 = (const float*)d_in[5];
    float* out = (float*)d_out;

    float* deg = (float*)d_ws;                      // N
    float* xw  = deg + N_NODES;                     // N*16
    float* agg = xw + (size_t)N_NODES * HID;        // N*16

    const int* row = ei;             // edge_index[0]
    const int* col = ei + N_EDGES;   // edge_index[1]

    gcn_init<<<(N_NODES * HID + 255) / 256, 256, 0, stream>>>(deg, agg);
    gcn_degree<<<(N_EDGES + 255) / 256, 256, 0, stream>>>(col, deg);
    gcn_rsqrt<<<(N_NODES + 255) / 256, 256, 0, stream>>>(deg);

    int n_tiles = N_NODES / 16;                                   // 6250
    gcn_xw_wmma<<<(n_tiles + 7) / 8, 256, 0, stream>>>(x, W1, xw);

    long long sthreads = (long long)N_EDGES * 16;
    gcn_scatter<<<(int)((sthreads + 255) / 256), 256, 0, stream>>>(row, col, deg, xw, agg);

    gcn_out<<<(N_NODES + 255) / 256, 256, 0, stream>>>(agg, xw, deg, b1, W2, b2, out);
}